// Attention_3015067042098
// MI455X (gfx1250) — compile-verified
//
#include <hip/hip_runtime.h>
#include <math.h>

// ---------------------------------------------------------------------------
// MI455X (gfx1250) implementation. wave32. All big GEMMs use
// V_WMMA_F32_16X16X4_F32 (exact fp32 matrix pipe). Problem is memory bound
// (~0.7GB traffic vs ~15 GFLOP), so fp32 WMMA + fusion of the final MLP chain
// is the right operating point.
// ---------------------------------------------------------------------------

#define DIM 128
#define NB  4
#define LL  16384   // H*W
#define C3  (3*DIM)

typedef __attribute__((ext_vector_type(2))) float v2f;
typedef __attribute__((ext_vector_type(8))) float v8f;

__device__ __constant__ int d_G[4]    = {16,32,32,48};
__device__ __constant__ int d_OFF[4]  = {0,16,48,80};
__device__ __constant__ int d_GOFF[4] = {0,256,1280,2304};   // prefix of g*g
__device__ __constant__ int d_segGrp[8] = {0,1,1,2,2,3,3,3}; // out-gemm i-tile map
__device__ __constant__ int d_segIt[8]  = {0,0,1,0,1,0,1,2};

__device__ __forceinline__ v8f wmma4(v2f a, v2f b, v8f c) {
  // D(16x16) = A(16x4) * B(4x16) + C, fp32
  return __builtin_amdgcn_wmma_f32_16x16x4_f32(false, a, false, b,
                                               (short)0, c, false, false);
}
__device__ __forceinline__ float gelu_exact(float x) {
  return 0.5f * x * (1.0f + erff(x * 0.70710678118654752f));
}

// ===========================================================================
// K1: qkv = qkv_w(384x128) @ x(b,128,L) -> (b,384,L). One wave per 16x16 tile.
// A layout (16x4 f32): lanes0-15 hold M=lane, K=0(v.x),1(v.y); lanes16-31 K=2,3
// B layout (4x16):     v.x = B[2*half][r], v.y = B[2*half+1][r]
// D layout: vgpr q -> (M = q + 8*half, N = lane&15)
// ===========================================================================
__global__ __launch_bounds__(256) void k_qkv(const float* __restrict__ W,
                                             const float* __restrict__ x,
                                             float* __restrict__ out) {
  int gid  = blockIdx.x * 256 + threadIdx.x;
  int wave = gid >> 5, lane = gid & 31;
  int lt = wave & 1023;
  int ot = (wave >> 10) % 24;
  int b  = wave / (1024 * 24);
  int m0 = ot << 4, n0 = lt << 4;
  int half = lane >> 4, r = lane & 15;
  const float* xb = x + (size_t)b * DIM * LL;
  v8f acc = {0,0,0,0,0,0,0,0};
  for (int k0 = 0; k0 < DIM; k0 += 4) {
    v2f a, bv;
    a.x  = W[(m0 + r) * DIM + k0 + 2*half];
    a.y  = W[(m0 + r) * DIM + k0 + 2*half + 1];
    bv.x = xb[(size_t)(k0 + 2*half)     * LL + n0 + r];
    bv.y = xb[(size_t)(k0 + 2*half + 1) * LL + n0 + r];
    acc = wmma4(a, bv, acc);
  }
  float* ob = out + (size_t)b * C3 * LL;
  #pragma unroll
  for (int q = 0; q < 8; ++q)
    ob[(size_t)(m0 + q + 8*half) * LL + n0 + r] = acc[q];
}

// ===========================================================================
// K2: depthwise 3x3, pad 1 (cross-correlation, NCHW / OIHW)
// ===========================================================================
__global__ __launch_bounds__(256) void k_dwconv(const float* __restrict__ in,
                                                const float* __restrict__ w,
                                                float* __restrict__ out) {
  size_t i = (size_t)blockIdx.x * 256 + threadIdx.x;
  if (i >= (size_t)NB * C3 * LL) return;
  int col = (int)(i & 127);
  int row = (int)((i >> 7) & 127);
  int c   = (int)((i >> 14) % C3);
  const float* wp    = w + c * 9;
  const float* plane = in + (i & ~(size_t)(LL - 1));
  float s = 0.f;
  #pragma unroll
  for (int ky = 0; ky < 3; ++ky) {
    int y = row + ky - 1;
    if ((unsigned)y < 128u) {
      #pragma unroll
      for (int kx = 0; kx < 3; ++kx) {
        int xx = col + kx - 1;
        if ((unsigned)xx < 128u) s += wp[ky*3 + kx] * plane[(y << 7) + xx];
      }
    }
  }
  out[i] = s;
}

// ===========================================================================
// K3: qmean[c,l] = mean_b q[b,c,l]   (q = channels 0..127 of post-conv qkv)
// ===========================================================================
__global__ __launch_bounds__(256) void k_qmean(const float* __restrict__ qkv1,
                                               float* __restrict__ qm) {
  size_t i = (size_t)blockIdx.x * 256 + threadIdx.x;
  if (i >= (size_t)DIM * LL) return;
  int c = (int)(i / LL); size_t l = i % LL;
  float s = 0.f;
  for (int b = 0; b < NB; ++b) s += qkv1[((size_t)b * C3 + c) * LL + l];
  qm[i] = s * 0.25f;
}

__global__ __launch_bounds__(256) void k_mu(const float* __restrict__ qm,
                                            float* __restrict__ mu) {
  __shared__ float red[256];
  int c = blockIdx.x;
  float s = 0.f;
  for (int l = threadIdx.x; l < LL; l += 256) s += qm[(size_t)c * LL + l];
  red[threadIdx.x] = s; __syncthreads();
  for (int o = 128; o > 0; o >>= 1) {
    if (threadIdx.x < o) red[threadIdx.x] += red[threadIdx.x + o];
    __syncthreads();
  }
  if (threadIdx.x == 0) mu[c] = red[0] * (1.0f / LL);
}

// Gram = Qm Qm^T via WMMA, split-K (64 splits of 256 columns)
__global__ __launch_bounds__(256) void k_gram(const float* __restrict__ qm,
                                              float* __restrict__ part) {
  int gid = blockIdx.x * 256 + threadIdx.x;
  int wave = gid >> 5, lane = gid & 31;
  int ks = wave & 63, tile = wave >> 6;       // tile: 0..63 = it*8+jt
  int it = tile >> 3, jt = tile & 7;
  int half = lane >> 4, r = lane & 15;
  int l0 = ks * 256;
  v8f acc = {0,0,0,0,0,0,0,0};
  for (int k = l0; k < l0 + 256; k += 4) {
    v2f a, bv;
    a.x  = qm[(size_t)(it*16 + r) * LL + k + 2*half];
    a.y  = qm[(size_t)(it*16 + r) * LL + k + 2*half + 1];
    bv.x = qm[(size_t)(jt*16 + r) * LL + k + 2*half];
    bv.y = qm[(size_t)(jt*16 + r) * LL + k + 2*half + 1];
    acc = wmma4(a, bv, acc);
  }
  float* p = part + ((size_t)tile * 64 + ks) * 256;
  #pragma unroll
  for (int q = 0; q < 8; ++q) p[q * 32 + lane] = acc[q];
}

__global__ __launch_bounds__(256) void k_gram_red(const float* __restrict__ part,
                                                  const float* __restrict__ mu,
                                                  float* __restrict__ cov) {
  int tile = blockIdx.x, t = threadIdx.x;
  float s = 0.f;
  for (int ks = 0; ks < 64; ++ks) s += part[((size_t)tile * 64 + ks) * 256 + t];
  int q = t >> 5, lane = t & 31, half = lane >> 4, r = lane & 15;
  int row = (tile >> 3) * 16 + q + 8*half;
  int col = (tile & 7)  * 16 + r;
  cov[row * DIM + col] = s - (float)LL * mu[row] * mu[col];
}

// corr row-means + stable descending argsort (matches jnp.argsort(-score))
__global__ __launch_bounds__(128) void k_rank(const float* __restrict__ cov,
                                              int* __restrict__ idx) {
  __shared__ float sc[128];
  int i = threadIdx.x;
  float dii = cov[i * DIM + i];
  float s = 0.f;
  for (int j = 0; j < DIM; ++j) {
    float c = cov[i * DIM + j] / sqrtf(dii * cov[j * DIM + j]);
    c = fminf(1.f, fmaxf(-1.f, c));
    s += c;
  }
  sc[i] = s;
  __syncthreads();
  if (i == 0) {
    bool used[128];
    for (int j = 0; j < 128; ++j) used[j] = false;
    for (int rk = 0; rk < 128; ++rk) {
      int best = 0; float bv = -3.0e38f;
      for (int j = 0; j < 128; ++j)
        if (!used[j] && sc[j] > bv) { bv = sc[j]; best = j; }
      used[best] = true;
      idx[rk] = best;
    }
  }
}

// ===========================================================================
// K4: permuted + L2-normalized q,k rows: qn[b,c,:] = q[b,idx[c],:]/||.||
// ===========================================================================
__global__ __launch_bounds__(256) void k_normperm(const float* __restrict__ qkv1,
                                                  const int* __restrict__ idx,
                                                  float* __restrict__ qn,
                                                  float* __restrict__ kn) {
  __shared__ float red[256];
  int bc = blockIdx.x; int b = bc >> 7; int c = bc & 127;
  int ch = idx[c];
  for (int which = 0; which < 2; ++which) {
    const float* src = qkv1 + ((size_t)b * C3 + which * DIM + ch) * LL;
    float* dst = (which ? kn : qn) + ((size_t)b * DIM + c) * LL;
    float s = 0.f;
    for (int l = threadIdx.x; l < LL; l += 256) { float v = src[l]; s += v * v; }
    red[threadIdx.x] = s; __syncthreads();
    for (int o = 128; o > 0; o >>= 1) {
      if (threadIdx.x < o) red[threadIdx.x] += red[threadIdx.x + o];
      __syncthreads();
    }
    float inv = 1.0f / fmaxf(sqrtf(red[0]), 1e-12f);
    for (int l = threadIdx.x; l < LL; l += 256) dst[l] = src[l] * inv;
    __syncthreads();
  }
}

// ===========================================================================
// K5: aligned_g = P * cache * P^T (block-average pooling), per group
// ===========================================================================
__global__ __launch_bounds__(256) void k_aligned(const float* __restrict__ cache,
                                                 float* __restrict__ alg) {
  int grp = blockIdx.x;
  int g = d_G[grp];
  float* out = alg + d_GOFF[grp];
  for (int t = threadIdx.x; t < g * g; t += 256) {
    int i = t / g, j = t % g;
    int rs = i * DIM / g, re = ((i + 1) * DIM + g - 1) / g;
    int cs = j * DIM / g, ce = ((j + 1) * DIM + g - 1) / g;
    float s = 0.f;
    for (int rr = rs; rr < re; ++rr)
      for (int cc = cs; cc < ce; ++cc) s += cache[rr * DIM + cc];
    out[i * g + j] = s / (float)((re - rs) * (ce - cs));
  }
}

// ===========================================================================
// K6: attention logits qn*kn^T per group, split-K WMMA (K = L = 16384)
// ===========================================================================
__device__ __forceinline__ void tile_decode(int tid, int& grp, int& it, int& jt) {
  if (tid < 1)       { grp = 0; it = 0; jt = 0; }
  else if (tid < 5)  { int l = tid - 1; grp = 1; it = l >> 1; jt = l & 1; }
  else if (tid < 9)  { int l = tid - 5; grp = 2; it = l >> 1; jt = l & 1; }
  else               { int l = tid - 9; grp = 3; it = l / 3;  jt = l % 3; }
}

__global__ __launch_bounds__(256) void k_logits(const float* __restrict__ qn,
                                                const float* __restrict__ kn,
                                                float* __restrict__ part) {
  int gid = blockIdx.x * 256 + threadIdx.x;
  int wave = gid >> 5, lane = gid & 31;
  int ks = wave & 63;
  int bt = wave >> 6;              // 0..71 = b*18 + tid
  int b = bt / 18, tid = bt % 18;
  int grp, it, jt; tile_decode(tid, grp, it, jt);
  int off = d_OFF[grp];
  int half = lane >> 4, r = lane & 15;
  const float* qa = qn + ((size_t)b * DIM + off + it * 16 + r) * LL;
  const float* kb = kn + ((size_t)b * DIM + off + jt * 16 + r) * LL;
  int l0 = ks * 256;
  v8f acc = {0,0,0,0,0,0,0,0};
  for (int k = l0; k < l0 + 256; k += 4) {
    v2f a, bv;
    a.x  = qa[k + 2*half];  a.y  = qa[k + 2*half + 1];
    bv.x = kb[k + 2*half];  bv.y = kb[k + 2*half + 1];
    acc = wmma4(a, bv, acc);
  }
  float* p = part + ((size_t)bt * 64 + ks) * 256;
  #pragma unroll
  for (int q = 0; q < 8; ++q) p[q * 32 + lane] = acc[q];
}

__global__ __launch_bounds__(256) void k_logits_red(const float* __restrict__ part,
                                                    const float* __restrict__ temp,
                                                    float* __restrict__ logits) {
  int bt = blockIdx.x, t = threadIdx.x;
  int b = bt / 18, tid = bt % 18;
  int grp, it, jt; tile_decode(tid, grp, it, jt);
  int g = d_G[grp];
  float s = 0.f;
  for (int ks = 0; ks < 64; ++ks) s += part[((size_t)bt * 64 + ks) * 256 + t];
  int q = t >> 5, lane = t & 31, half = lane >> 4, r = lane & 15;
  int row = it * 16 + q + 8*half, col = jt * 16 + r;
  logits[(size_t)b * 4608 + d_GOFF[grp] + row * g + col] = s * temp[grp];
}

// ===========================================================================
// K7: per-(b,group) chain: softmax -> pw=cw(aligned+a) -> x1=a*scale+shift
//     -> x1*gelu(gate(x1)) -> a += ...   (all in LDS; g <= 48)
// ===========================================================================
struct InterP {
  const float* cw_w[4]; const float* cw_b[4];
  const float* gw[4];   const float* gb[4];
};

__global__ __launch_bounds__(256) void k_groupchain(const float* __restrict__ logits,
                                                    const float* __restrict__ alg,
                                                    InterP P,
                                                    float* __restrict__ att) {
  __shared__ float sa[2304];   // a (g*g)
  __shared__ float sx[2304];   // x1
  __shared__ float spw[4608];  // pw (2g*g), later u (g*g)
  int b = blockIdx.x >> 2, grp = blockIdx.x & 3;
  int g = d_G[grp], gg = g * g;
  const float* lg = logits + (size_t)b * 4608 + d_GOFF[grp];
  const float* al = alg + d_GOFF[grp];
  for (int t = threadIdx.x; t < gg; t += 256) sa[t] = lg[t];
  __syncthreads();
  if (threadIdx.x < (unsigned)g) {                     // softmax row
    float* row = sa + threadIdx.x * g;
    float m = -3.0e38f;
    for (int j = 0; j < g; ++j) m = fmaxf(m, row[j]);
    float s = 0.f;
    for (int j = 0; j < g; ++j) { row[j] = expf(row[j] - m); s += row[j]; }
    float inv = 1.0f / s;
    for (int j = 0; j < g; ++j) row[j] *= inv;
  }
  __syncthreads();
  for (int t = threadIdx.x; t < 2 * gg; t += 256) {    // pw = cw*(aligned+a)+b
    int o = t / g, j = t % g;
    float s = P.cw_b[grp][o];
    for (int i = 0; i < g; ++i)
      s += P.cw_w[grp][o * g + i] * (al[i * g + j] + sa[i * g + j]);
    spw[t] = s;
  }
  __syncthreads();
  for (int t = threadIdx.x; t < gg; t += 256)          // x1 = a*scale + shift
    sx[t] = sa[t] * spw[t] + spw[gg + t];
  __syncthreads();
  for (int t = threadIdx.x; t < gg; t += 256) {        // u = gate(x1)
    int o = t / g, j = t % g;
    float s = P.gb[grp][o];
    for (int i = 0; i < g; ++i) s += P.gw[grp][o * g + i] * sx[i * g + j];
    spw[t] = s;
  }
  __syncthreads();
  float* ao = att + (size_t)b * 4608 + d_GOFF[grp];
  for (int t = threadIdx.x; t < gg; t += 256)
    ao[t] = sx[t] * gelu_exact(spw[t]) + sa[t];
}

// ===========================================================================
// K8: out_all[b,off+i,l] = sum_h a[b,i,h] * v[b, idx[off+h], l]  (WMMA, K=g)
// ===========================================================================
__global__ __launch_bounds__(256) void k_outgemm(const float* __restrict__ att,
                                                 const float* __restrict__ qkv1,
                                                 const int* __restrict__ idx,
                                                 float* __restrict__ oall) {
  int gid = blockIdx.x * 256 + threadIdx.x;
  int wave = gid >> 5, lane = gid & 31;
  int lt = wave & 1023;
  int rest = wave >> 10;          // 0..31
  int b = rest >> 3, seg = rest & 7;
  int grp = d_segGrp[seg], it = d_segIt[seg];
  int g = d_G[grp], off = d_OFF[grp];
  int n0 = lt << 4;
  int half = lane >> 4, r = lane & 15;
  const float* ab = att + (size_t)b * 4608 + d_GOFF[grp];
  const float* vb = qkv1 + ((size_t)b * C3 + 2 * DIM) * LL;
  v8f acc = {0,0,0,0,0,0,0,0};
  for (int h0 = 0; h0 < g; h0 += 4) {
    v2f a, bv;
    a.x = ab[(it * 16 + r) * g + h0 + 2*half];
    a.y = ab[(it * 16 + r) * g + h0 + 2*half + 1];
    int c0 = idx[off + h0 + 2*half];
    int c1 = idx[off + h0 + 2*half + 1];
    bv.x = vb[(size_t)c0 * LL + n0 + r];
    bv.y = vb[(size_t)c1 * LL + n0 + r];
    acc = wmma4(a, bv, acc);
  }
  #pragma unroll
  for (int q = 0; q < 8; ++q)
    oall[((size_t)b * DIM + off + it * 16 + q + 8*half) * LL + n0 + r] = acc[q];
}

// ===========================================================================
// K9: new_cache = 0.9*cache + 0.1 * sum_g floor(bilinear(att_g.mean(0))*g/128)
// ===========================================================================
__global__ __launch_bounds__(256) void k_cache(const float* __restrict__ att,
                                               const float* __restrict__ cache,
                                               float* __restrict__ out2) {
  int t = blockIdx.x * 256 + threadIdx.x;
  int c0 = t >> 7, c1 = t & 127;
  float ue = 0.f;
  for (int grp = 0; grp < 4; ++grp) {
    int g = d_G[grp], go = d_GOFF[grp];
    float fy = (c0 + 0.5f) * (float)g / 128.f - 0.5f;
    float fx = (c1 + 0.5f) * (float)g / 128.f - 0.5f;
    int y0 = (int)floorf(fy), x0 = (int)floorf(fx);
    float ty = fy - y0, tx = fx - x0;
    int y0c = min(max(y0, 0), g - 1), y1c = min(max(y0 + 1, 0), g - 1);
    int x0c = min(max(x0, 0), g - 1), x1c = min(max(x0 + 1, 0), g - 1);
    float m00 = 0.f, m01 = 0.f, m10 = 0.f, m11 = 0.f;
    for (int b = 0; b < NB; ++b) {
      const float* ab = att + (size_t)b * 4608 + go;
      m00 += ab[y0c * g + x0c]; m01 += ab[y0c * g + x1c];
      m10 += ab[y1c * g + x0c]; m11 += ab[y1c * g + x1c];
    }
    float v = 0.25f * ((1.f - ty) * ((1.f - tx) * m00 + tx * m01) +
                       ty         * ((1.f - tx) * m10 + tx * m11));
    ue += floorf(v * (float)g / 128.f);
  }
  out2[t] = cache[t] * 0.9f + ue * 0.1f;
}

// ===========================================================================
// K10: fused intra-MLP + proj, one wave per 16-column tile:
//   s = qn+kn+out_all ; xg = gelu(Wg s + bg) * s ; t1 = Wd xg + bd ;
//   o = Wu t1 + bu + out_all ; out = Wp o
// LDS: 2 x (128x16) fp32 buffers per wave (16KB/wave, 64KB/WG)
// ===========================================================================
__global__ __launch_bounds__(128) void k_mlp(const float* __restrict__ qn,
                                             const float* __restrict__ kn,
                                             const float* __restrict__ oall,
                                             const float* __restrict__ gw,
                                             const float* __restrict__ gb,
                                             const float* __restrict__ dw,
                                             const float* __restrict__ db,
                                             const float* __restrict__ uw,
                                             const float* __restrict__ ub,
                                             const float* __restrict__ pw,
                                             float* __restrict__ out) {
  __shared__ float lds[4][2][128 * 16];
  int wid = threadIdx.x >> 5, lane = threadIdx.x & 31;
  int w = blockIdx.x * 4 + wid;
  int lt = w & 1023, b = w >> 10;
  int l0 = lt << 4;
  float* sb = lds[wid][0];
  float* tb = lds[wid][1];
  int half = lane >> 4, r = lane & 15;

  // stage 1: s tile -> sb
  for (int e = lane; e < 2048; e += 32) {
    int c = e >> 4, col = e & 15;
    size_t gi = ((size_t)b * DIM + c) * LL + l0 + col;
    sb[e] = qn[gi] + kn[gi] + oall[gi];
  }
  // stage 2: xg = gelu(Wg s + bg) * s -> tb
  for (int mt = 0; mt < 8; ++mt) {
    v8f acc = {0,0,0,0,0,0,0,0};
    for (int k0 = 0; k0 < 128; k0 += 4) {
      v2f a, bv;
      a.x  = gw[(mt * 16 + r) * 128 + k0 + 2*half];
      a.y  = gw[(mt * 16 + r) * 128 + k0 + 2*half + 1];
      bv.x = sb[(k0 + 2*half)     * 16 + r];
      bv.y = sb[(k0 + 2*half + 1) * 16 + r];
      acc = wmma4(a, bv, acc);
    }
    #pragma unroll
    for (int q = 0; q < 8; ++q) {
      int row = mt * 16 + q + 8*half;
      float u = acc[q] + gb[row];
      tb[row * 16 + r] = sb[row * 16 + r] * gelu_exact(u);
    }
  }
  // stage 3: t1 = Wd xg + bd -> sb rows 0..63
  for (int mt = 0; mt < 4; ++mt) {
    v8f acc = {0,0,0,0,0,0,0,0};
    for (int k0 = 0; k0 < 128; k0 += 4) {
      v2f a, bv;
      a.x  = dw[(mt * 16 + r) * 128 + k0 + 2*half];
      a.y  = dw[(mt * 16 + r) * 128 + k0 + 2*half + 1];
      bv.x = tb[(k0 + 2*half)     * 16 + r];
      bv.y = tb[(k0 + 2*half + 1) * 16 + r];
      acc = wmma4(a, bv, acc);
    }
    #pragma unroll
    for (int q = 0; q < 8; ++q) {
      int row = mt * 16 + q + 8*half;
      sb[row * 16 + r] = acc[q] + db[row];
    }
  }
  // stage 4: o = Wu t1 + bu + out_all -> tb
  for (int mt = 0; mt < 8; ++mt) {
    v8f acc = {0,0,0,0,0,0,0,0};
    for (int k0 = 0; k0 < 64; k0 += 4) {
      v2f a, bv;
      a.x  = uw[(mt * 16 + r) * 64 + k0 + 2*half];
      a.y  = uw[(mt * 16 + r) * 64 + k0 + 2*half + 1];
      bv.x = sb[(k0 + 2*half)     * 16 + r];
      bv.y = sb[(k0 + 2*half + 1) * 16 + r];
      acc = wmma4(a, bv, acc);
    }
    #pragma unroll
    for (int q = 0; q < 8; ++q) {
      int row = mt * 16 + q + 8*half;
      tb[row * 16 + r] = acc[q] + ub[row] +
                         oall[((size_t)b * DIM + row) * LL + l0 + r];
    }
  }
  // stage 5: out = Wp o -> global
  for (int mt = 0; mt < 8; ++mt) {
    v8f acc = {0,0,0,0,0,0,0,0};
    for (int k0 = 0; k0 < 128; k0 += 4) {
      v2f a, bv;
      a.x  = pw[(mt * 16 + r) * 128 + k0 + 2*half];
      a.y  = pw[(mt * 16 + r) * 128 + k0 + 2*half + 1];
      bv.x = tb[(k0 + 2*half)     * 16 + r];
      bv.y = tb[(k0 + 2*half + 1) * 16 + r];
      acc = wmma4(a, bv, acc);
    }
    #pragma unroll
    for (int q = 0; q < 8; ++q) {
      int row = mt * 16 + q + 8*half;
      out[((size_t)b * DIM + row) * LL + l0 + r] = acc[q];
    }
  }
}

// ===========================================================================
// Host launcher
// ===========================================================================
extern "C" void kernel_launch(void* const* d_in, const int* in_sizes, int n_in,
                              void* d_out, int out_size, void* d_ws, size_t ws_size,
                              hipStream_t stream) {
  (void)in_sizes; (void)n_in; (void)out_size; (void)ws_size;
  // Flattened input order: top-level dict insertion order (x, qv_cache, params),
  // params flattened as a JAX pytree (sorted keys):
  //   dw_w, inter[0..3]{cw_b,cw_w,gate_b,gate_w}, intra{down_b,down_w,gate_b,
  //   gate_w,up_b,up_w}, proj_w, qkv_w, temperature
  const float* x      = (const float*)d_in[0];
  const float* cache  = (const float*)d_in[1];
  const float* dw_w   = (const float*)d_in[2];
  InterP P;
  for (int g = 0; g < 4; ++g) {
    P.cw_b[g] = (const float*)d_in[3 + 4*g + 0];
    P.cw_w[g] = (const float*)d_in[3 + 4*g + 1];
    P.gb[g]   = (const float*)d_in[3 + 4*g + 2];
    P.gw[g]   = (const float*)d_in[3 + 4*g + 3];
  }
  const float* i_db   = (const float*)d_in[19];
  const float* i_dw   = (const float*)d_in[20];
  const float* i_gb   = (const float*)d_in[21];
  const float* i_gw   = (const float*)d_in[22];
  const float* i_ub   = (const float*)d_in[23];
  const float* i_uw   = (const float*)d_in[24];
  const float* proj_w = (const float*)d_in[25];
  const float* qkv_w  = (const float*)d_in[26];
  const float* temp   = (const float*)d_in[27];

  float* out  = (float*)d_out;                 // (4,128,128,128)
  float* out2 = out + (size_t)NB * DIM * LL;   // new_cache (128,128)

  float* ws   = (float*)d_ws;
  float* qkv0 = ws;                            // 25165824
  float* qkv1 = ws + 25165824;                 // 25165824
  float* qn   = ws + 50331648;                 // 8388608
  float* kn   = ws + 58720256;                 // 8388608
  float* oall = ws + 67108864;                 // 8388608
  // small buffers aliased over qkv0 (dead after k_dwconv)
  float* qm    = qkv0;                         // 2097152
  float* mu    = qkv0 + 2097152;               // 128
  float* cov   = qkv0 + 2097280;               // 16384
  int*   idx   = (int*)(qkv0 + 2113664);       // 128
  float* alg   = qkv0 + 2113792;               // 4608
  float* logit = qkv0 + 2118400;               // 18432
  float* attb  = qkv0 + 2136832;               // 18432
  float* part  = qkv0 + 2155264;               // 1179648

  k_qkv       <<<12288, 256, 0, stream>>>(qkv_w, x, qkv0);
  k_dwconv    <<<98304, 256, 0, stream>>>(qkv0, dw_w, qkv1);
  k_qmean     <<<8192,  256, 0, stream>>>(qkv1, qm);
  k_mu        <<<128,   256, 0, stream>>>(qm, mu);
  k_gram      <<<512,   256, 0, stream>>>(qm, part);
  k_gram_red  <<<64,    256, 0, stream>>>(part, mu, cov);
  k_rank      <<<1,     128, 0, stream>>>(cov, idx);
  k_normperm  <<<512,   256, 0, stream>>>(qkv1, idx, qn, kn);
  k_aligned   <<<4,     256, 0, stream>>>(cache, alg);
  k_logits    <<<576,   256, 0, stream>>>(qn, kn, part);
  k_logits_red<<<72,    256, 0, stream>>>(part, temp, logit);
  k_groupchain<<<16,    256, 0, stream>>>(logit, alg, P, attb);
  k_outgemm   <<<4096,  256, 0, stream>>>(attb, qkv1, idx, oall);
  k_cache     <<<64,    256, 0, stream>>>(attb, cache, out2);
  k_mlp       <<<1024,  128, 0, stream>>>(qn, kn, oall, i_gw, i_gb, i_dw, i_db,
                                          i_uw, i_ub, proj_w, out);
}